// LinearSplineLayer_23313082482867
// MI455X (gfx1250) — compile-verified
//
#include <hip/hip_runtime.h>
#include <stdint.h>

// Linear spline (B=262144, D=64, K=32): out[r] = (cdf_lo + alpha*pdf[idx]) scaled,
// pdf = softmax(params[r,:]). Memory-bound: 2.28 GB / 23.3 TB/s ~ 98 us.
// Strategy: wave-private double-buffered async global->LDS pipeline
// (global_load_async_to_lds_b128 th:TH_LOAD_NT + s_wait_asynccnt),
// coalesced 512B/instr copies, per-thread row compute via ds_load_b128.
// All global traffic is read-once/write-once -> non-temporal hints so the
// 2.15 GB param stream doesn't thrash the 192 MB L2.

#define WAVES_PER_BLOCK 8
#define BLOCK_THREADS   (WAVES_PER_BLOCK * 32)
#define KBINS           32
#define TILE_ROWS       32                      // rows per wave per iteration
#define TILE_FLOATS     (TILE_ROWS * KBINS)     // 1024 floats
#define TILE_BYTES      (TILE_FLOATS * 4)       // 4096 bytes

__global__ __launch_bounds__(BLOCK_THREADS)
void spline_async_kernel(const float* __restrict__ x,
                         const float* __restrict__ params,
                         float* __restrict__ out,
                         int nTiles)
{
    // 8 waves x 2 buffers x 4KB = 64 KB LDS (WGP has 320 KB -> 5 blocks resident)
    __shared__ float smem[WAVES_PER_BLOCK * 2 * TILE_FLOATS];

    const int lane = threadIdx.x & 31;
    const int wid  = threadIdx.x >> 5;

    float* wbuf0 = &smem[wid * 2 * TILE_FLOATS];
    float* wbuf1 = wbuf0 + TILE_FLOATS;
    // Low 32 bits of a flat LDS pointer are the LDS byte offset (aperture layout).
    const uint32_t lds0 = (uint32_t)(uintptr_t)wbuf0 + (uint32_t)(lane * 16);
    const uint32_t lds1 = (uint32_t)(uintptr_t)wbuf1 + (uint32_t)(lane * 16);

    const int waveGlobal = blockIdx.x * WAVES_PER_BLOCK + wid;
    const int waveCount  = gridDim.x * WAVES_PER_BLOCK;

    // Issue one 4KB tile copy: 8 x (32 lanes x b128) = 8 x 512B coalesced.
    // NT hint: stream is read exactly once, keep it out of the caches.
    auto issue_tile = [&](int tile, uint32_t ldsBase) {
        uint64_t ga = (uint64_t)(uintptr_t)params
                    + (uint64_t)tile * TILE_BYTES
                    + (uint64_t)(lane * 16);
        uint32_t la = ldsBase;
        #pragma unroll
        for (int j = 0; j < 8; ++j) {
            asm volatile("global_load_async_to_lds_b128 %0, %1, off th:TH_LOAD_NT"
                         :: "v"(la), "v"(ga) : "memory");
            la += 512;
            ga += 512;
        }
    };

    int tile = waveGlobal;
    if (tile < nTiles) issue_tile(tile, lds0);

    int cur = 0;
    for (; tile < nTiles; tile += waveCount) {
        const int nxt = tile + waveCount;
        if (nxt < nTiles) {
            // Prefetch next tile into the other buffer, then wait for the
            // older 8 async loads (in-order completion) of the current tile.
            issue_tile(nxt, cur ? lds0 : lds1);
            asm volatile("s_wait_asynccnt 8" ::: "memory");
        } else {
            asm volatile("s_wait_asynccnt 0" ::: "memory");
        }

        // This thread's row lives at a 16B-aligned LDS address.
        const float* rowp = (cur ? wbuf1 : wbuf0) + lane * KBINS;

        float v[KBINS];
        #pragma unroll
        for (int j = 0; j < 8; ++j) {            // ds_load_b128 x 8
            float4 q = ((const float4*)rowp)[j];
            v[4 * j + 0] = q.x;
            v[4 * j + 1] = q.y;
            v[4 * j + 2] = q.z;
            v[4 * j + 3] = q.w;
        }

        float m = v[0];
        #pragma unroll
        for (int k = 1; k < KBINS; ++k) m = fmaxf(m, v[k]);

        const int row = tile * TILE_ROWS + lane;
        const float xv = __builtin_nontemporal_load(&x[row]);

        // t in [0,1]; bin index and in-bin position
        float t01 = (xv + 1.1f) * (1.0f / 2.2f);
        t01 = fminf(fmaxf(t01, 0.0f), 1.0f);
        const float tk = t01 * (float)KBINS;
        int bi = (int)tk;                         // floor, t01 >= 0
        bi = (bi > KBINS - 1) ? (KBINS - 1) : bi;
        const float alpha = tk - (float)bi;

        float total = 0.0f, part = 0.0f, pb = 0.0f;
        #pragma unroll
        for (int k = 0; k < KBINS; ++k) {
            const float e = __expf(v[k] - m);     // v_exp_f32
            total += e;
            part  += (k < bi) ? e : 0.0f;         // exclusive prefix at bi
            pb     = (k == bi) ? e : pb;
        }

        const float y01 = (part + alpha * pb) / total;
        __builtin_nontemporal_store(y01 * 2.2f - 1.1f, &out[row]);

        cur ^= 1;
    }
}

extern "C" void kernel_launch(void* const* d_in, const int* in_sizes, int n_in,
                              void* d_out, int out_size, void* d_ws, size_t ws_size,
                              hipStream_t stream)
{
    const float* x      = (const float*)d_in[0];   // [B, D] f32
    const float* params = (const float*)d_in[1];   // [B, D, K] f32
    float* out          = (float*)d_out;           // [B, D] f32

    const int nRows  = in_sizes[0];                // B*D = 16,777,216
    const int nTiles = nRows / TILE_ROWS;          // 524,288

    // 4096 blocks x 8 waves = 32768 waves -> 16 tiles/wave: enough iterations
    // for the double-buffered async pipeline to amortize, enough blocks to
    // fill every WGP several times over.
    const int blocks = 4096;
    hipLaunchKernelGGL(spline_async_kernel, dim3(blocks), dim3(BLOCK_THREADS),
                       0, stream, x, params, out, nTiles);
}